// EProPnPLoss_29025388986378
// MI455X (gfx1250) — compile-verified
//
#include <hip/hip_runtime.h>
#include <math.h>

// ---------------- problem constants (match reference) ----------------
#define Bn   8
#define Sn   4
#define Nq   1024
#define PSQ  256          // 16*16 elements per patch
#define Himg 256
#define Wimg 256
#define EPSB 1e-4f

typedef __attribute__((ext_vector_type(2))) float v2f;
typedef __attribute__((ext_vector_type(8))) float v8f;

// =====================================================================
// Kernel 1: w_patch = mean( dense_weight / (max(dense_b,eps)+eps) )
// one wave per patch; float4 (b128) fully-coalesced loads; shuffle reduce.
// Also counts mask = valid & (w_patch>0) per (b,s) sample via atomics.
// =====================================================================
__global__ __launch_bounds__(256) void k_wpatch(const float* __restrict__ db,
                                                const float* __restrict__ dw,
                                                const int*   __restrict__ pc,
                                                float* __restrict__ w_eff,
                                                int*   __restrict__ cnt)
{
    int gpatch = blockIdx.x * 8 + (threadIdx.x >> 5);   // 32768 patches
    int lane   = threadIdx.x & 31;
    const float4* pb = (const float4*)(db + (size_t)gpatch * PSQ);
    const float4* pw = (const float4*)(dw + (size_t)gpatch * PSQ);
    float s = 0.f;
#pragma unroll
    for (int k = 0; k < 2; ++k) {                       // 2 x 32 lanes x float4 = 256
        float4 bv = pb[lane + 32 * k];
        float4 wv = pw[lane + 32 * k];
        s += wv.x / (fmaxf(bv.x, EPSB) + EPSB);
        s += wv.y / (fmaxf(bv.y, EPSB) + EPSB);
        s += wv.z / (fmaxf(bv.z, EPSB) + EPSB);
        s += wv.w / (fmaxf(bv.w, EPSB) + EPSB);
    }
#pragma unroll
    for (int off = 16; off; off >>= 1) s += __shfl_xor(s, off, 32);
    float wp = s * (1.0f / 256.0f);
    int cls = pc[gpatch];
    bool mask = (cls >= 0) && (cls < Nq) && (wp > 0.f);
    if (lane == 0) {
        w_eff[gpatch] = mask ? wp : 0.f;
        atomicAdd(&cnt[gpatch >> 10], mask ? 1 : 0);
    }
}

// =====================================================================
// Kernel 2: back-project buddy patch centers -> X_obj, pack (X,Y,Z,w)
// =====================================================================
__global__ __launch_bounds__(256) void k_xobj(const int*   __restrict__ pc,
                                              const float* __restrict__ K,
                                              const float* __restrict__ poses,
                                              const float* __restrict__ depth,
                                              const float* __restrict__ w_eff,
                                              float4* __restrict__ XW)
{
    int g  = blockIdx.x * 256 + threadIdx.x;  // 32768
    int bs = g >> 10;
    int b  = bs >> 2, s = bs & 3;

    const float* Kt = K + (size_t)(b * (Sn + 1) + (s + 1)) * 9;   // K[:,1:]
    float fx = Kt[0], cx = Kt[2], fy = Kt[4], cy = Kt[5];

    const float* T = poses + (size_t)(b * (Sn + 1) + (s + 1)) * 16; // poses[:,1:]
    float R00=T[0], R01=T[1], R02=T[2],  tx=T[3];
    float R10=T[4], R11=T[5], R12=T[6],  ty=T[7];
    float R20=T[8], R21=T[9], R22=T[10], tz=T[11];

    int cls   = pc[g];
    int buddy = min(max(cls, 0), Nq - 1);
    float cxj = ((float)(buddy & 31) + 0.5f) * 8.0f;   // xs[buddy%32]
    float cyj = ((float)(buddy >> 5) + 0.5f) * 8.0f;   // ys[buddy/32]
    int xi = min(max(__float2int_rn(cxj), 0), Wimg - 1);
    int yi = min(max(__float2int_rn(cyj), 0), Himg - 1);
    float d = depth[((size_t)bs * Himg + yi) * Wimg + xi];

    float Xc = (cxj - cx) / fx * d;
    float Yc = (cyj - cy) / fy * d;
    float Zc = d;
    // X_obj = R^T (X_cam - t)
    float dx = Xc - tx, dy = Yc - ty, dz = Zc - tz;
    float4 o;
    o.x = R00*dx + R10*dy + R20*dz;
    o.y = R01*dx + R11*dy + R21*dz;
    o.z = R02*dx + R12*dy + R22*dz;
    o.w = w_eff[g];
    XW[g] = o;
}

// =====================================================================
// Kernel 3: weighted 12x12 DLT Gram via V_WMMA_F32_16X16X4_F32.
// One block per (b,s) sample; 8 waves x 64 WMMA steps (4 A-rows = 2 points
// per step); branchless operand build (one-hot + range masks are loop-
// invariant and hoisted), ds_add_f32 cross-wave reduce in LDS.
// Gram(i,j) = sum_r w_r * a_r[i] * a_r[j]  (A operand pre-scaled by w).
// =====================================================================
__global__ __launch_bounds__(256) void k_gram(const float4* __restrict__ XW,
                                              float* __restrict__ Gw)
{
    __shared__ float4 sXW[Nq];
    __shared__ float  tile[256];
    int bs = blockIdx.x;
    int t  = threadIdx.x;

    const float4* src = XW + (size_t)bs * Nq;
#pragma unroll
    for (int i = t; i < Nq; i += 256) sXW[i] = src[i];
    tile[t] = 0.f;
    __syncthreads();

    int wave = t >> 5, lane = t & 31;
    int c    = lane & 15;        // Gram row/col index this lane feeds
    int half = lane >> 4;        // K-pair {0,1} vs {2,3} => point 2s / 2s+1

    // Loop-invariant branchless selectors:
    //   row element = base(comp) * mult, base = [X,Y,Z,1][c&3]
    //   u-row mult = cu - cm*u ; v-row mult = cv - cm*v
    int   comp = c & 3;
    float e0 = (comp == 0) ? 1.f : 0.f;
    float e1 = (comp == 1) ? 1.f : 0.f;
    float e2 = (comp == 2) ? 1.f : 0.f;
    float e3 = (comp == 3) ? 1.f : 0.f;
    float cu = (c < 4) ? 1.f : 0.f;
    float cv = (c >= 4 && c < 8) ? 1.f : 0.f;
    float cm = (c >= 8 && c < 12) ? 1.f : 0.f;

    v8f acc = {};
    int s0 = wave * 64;
#pragma unroll 4
    for (int st = s0; st < s0 + 64; ++st) {
        int p = 2 * st + half;                 // point index for this lane's rows
        float4 xw = sXW[p];
        float u = ((float)(p & 31) + 0.5f) * 8.0f;
        float v = ((float)(p >> 5) + 0.5f) * 8.0f;

        float base = e0 * xw.x + e1 * xw.y + e2 * xw.z + e3;
        float au = base * (cu - cm * u);       // u-row element, col c
        float av = base * (cv - cm * v);       // v-row element, col c

        v2f Am; Am.x = au * xw.w; Am.y = av * xw.w;   // A: 16x4, weighted
        v2f Bm; Bm.x = au;        Bm.y = av;          // B: 4x16, unweighted
        acc = __builtin_amdgcn_wmma_f32_16x16x4_f32(
                  false, Am, false, Bm, (short)0, acc, false, false);
    }

    // D layout: lane holds D[m][n], m = vgpr + 8*half, n = c
#pragma unroll
    for (int vr = 0; vr < 8; ++vr) {
        atomicAdd(&tile[(vr + 8 * half) * 16 + c], acc[vr]);
    }
    __syncthreads();
    Gw[(size_t)bs * 256 + t] = tile[t];
}

// =====================================================================
// Kernel 4: per-sample tail (1 wave, thread = sample):
//   solve (G+1e-6 I) p = rhs, P -> M = Kinv P, sign fix, polar-decompose
//   M3 (Jacobi on M3^T M3), R = M3 (M3^T M3)^(-1/2) = U V^T (d=+1 since
//   det>=0 after sign fix), t = M[:,3]/mean(sigma), errors, masked mean.
// =====================================================================
__global__ __launch_bounds__(32) void k_solve(const float* __restrict__ Gw,
                                              const int*   __restrict__ cnt,
                                              const float* __restrict__ K,
                                              const float* __restrict__ gt,
                                              float* __restrict__ out)
{
    int bs = threadIdx.x;             // 0..31
    int b  = bs >> 2;
    const float* G = Gw + (size_t)bs * 256;

    // ---- 11x11 solve, Gaussian elimination with partial pivoting ----
    float M[11][12];
    for (int i = 0; i < 11; ++i) {
        for (int j = 0; j < 11; ++j) M[i][j] = G[i * 16 + j] + ((i == j) ? 1e-6f : 0.f);
        M[i][11] = -G[i * 16 + 11];
    }
    for (int k = 0; k < 11; ++k) {
        int piv = k; float mx = fabsf(M[k][k]);
        for (int r = k + 1; r < 11; ++r) { float a = fabsf(M[r][k]); if (a > mx) { mx = a; piv = r; } }
        if (piv != k) for (int j = k; j < 12; ++j) { float tm = M[k][j]; M[k][j] = M[piv][j]; M[piv][j] = tm; }
        float inv = 1.0f / M[k][k];
        for (int r = k + 1; r < 11; ++r) {
            float f = M[r][k] * inv;
            for (int j = k; j < 12; ++j) M[r][j] -= f * M[k][j];
        }
    }
    float p[12];
    for (int i = 10; i >= 0; --i) {
        float a = M[i][11];
        for (int j = i + 1; j < 11; ++j) a -= M[i][j] * p[j];
        p[i] = a / M[i][i];
    }
    p[11] = 1.0f;

    // ---- M = Kinv @ P (Kq = K[b,0], analytic inverse) ----
    const float* Kq = K + (size_t)(b * (Sn + 1)) * 9;
    float fx = Kq[0], cx = Kq[2], fy = Kq[4], cy = Kq[5];
    float Mm[3][4];
    for (int j = 0; j < 4; ++j) {
        float P0 = p[j], P1 = p[4 + j], P2 = p[8 + j];
        Mm[0][j] = (P0 - cx * P2) / fx;
        Mm[1][j] = (P1 - cy * P2) / fy;
        Mm[2][j] = P2;
    }
    float det = Mm[0][0]*(Mm[1][1]*Mm[2][2]-Mm[1][2]*Mm[2][1])
              - Mm[0][1]*(Mm[1][0]*Mm[2][2]-Mm[1][2]*Mm[2][0])
              + Mm[0][2]*(Mm[1][0]*Mm[2][1]-Mm[1][1]*Mm[2][0]);
    float sg = (det < 0.f) ? -1.f : 1.f;
    for (int i = 0; i < 3; ++i) for (int j = 0; j < 4; ++j) Mm[i][j] *= sg;

    // ---- Jacobi eigendecomposition of S = M3^T M3 ----
    float A3[3][3];
    for (int i = 0; i < 3; ++i) for (int j = 0; j < 3; ++j) A3[i][j] = Mm[i][j];
    float S[3][3];
    for (int i = 0; i < 3; ++i)
        for (int j = 0; j < 3; ++j) {
            float a = 0.f;
            for (int k = 0; k < 3; ++k) a += A3[k][i] * A3[k][j];
            S[i][j] = a;
        }
    float V[3][3] = {{1,0,0},{0,1,0},{0,0,1}};
    for (int sweep = 0; sweep < 12; ++sweep) {
        for (int pr = 0; pr < 3; ++pr) {
            int pp = (pr == 2) ? 1 : 0;
            int qq = (pr == 0) ? 1 : 2;
            float apq = S[pp][qq];
            if (fabsf(apq) > 1e-20f) {
                float tau = (S[qq][qq] - S[pp][pp]) / (2.0f * apq);
                float tt  = copysignf(1.0f, tau) / (fabsf(tau) + sqrtf(1.0f + tau * tau));
                float cth = 1.0f / sqrtf(1.0f + tt * tt);
                float sth = tt * cth;
                for (int k = 0; k < 3; ++k) {
                    float a = S[k][pp], bvv = S[k][qq];
                    S[k][pp] = cth * a - sth * bvv;
                    S[k][qq] = sth * a + cth * bvv;
                }
                for (int k = 0; k < 3; ++k) {
                    float a = S[pp][k], bvv = S[qq][k];
                    S[pp][k] = cth * a - sth * bvv;
                    S[qq][k] = sth * a + cth * bvv;
                }
                for (int k = 0; k < 3; ++k) {
                    float a = V[k][pp], bvv = V[k][qq];
                    V[k][pp] = cth * a - sth * bvv;
                    V[k][qq] = sth * a + cth * bvv;
                }
            }
        }
    }
    float sig[3];
    for (int i = 0; i < 3; ++i) sig[i] = sqrtf(fmaxf(S[i][i], 1e-30f));

    // S^{-1/2} = V diag(1/sig) V^T ;  R = M3 * S^{-1/2}  (= U V^T, d=+1)
    float Sih[3][3];
    for (int i = 0; i < 3; ++i)
        for (int j = 0; j < 3; ++j) {
            float a = 0.f;
            for (int k = 0; k < 3; ++k) a += V[i][k] * V[j][k] / sig[k];
            Sih[i][j] = a;
        }
    float R[3][3];
    for (int i = 0; i < 3; ++i)
        for (int j = 0; j < 3; ++j) {
            float a = 0.f;
            for (int k = 0; k < 3; ++k) a += A3[i][k] * Sih[k][j];
            R[i][j] = a;
        }
    float msig = (sig[0] + sig[1] + sig[2]) * (1.0f / 3.0f);
    float tp0 = Mm[0][3] / msig, tp1 = Mm[1][3] / msig, tp2 = Mm[2][3] / msig;

    // ---- errors vs gt_pose ----
    const float* Tg = gt + (size_t)bs * 16;
    // trace(R_pred @ R_gt^T) = sum_ij Rp[i][j]*Rg[i][j]
    float tr = 0.f;
    for (int i = 0; i < 3; ++i)
        for (int j = 0; j < 3; ++j) tr += R[i][j] * Tg[i * 4 + j];
    float cosang = fminf(fmaxf((tr - 1.0f) * 0.5f, -1.0f + 1e-7f), 1.0f - 1e-7f);
    float rot_err = acosf(cosang);
    float d0 = tp0 - Tg[3], d1 = tp1 - Tg[7], d2 = tp2 - Tg[11];
    float trans_err = sqrtf(d0 * d0 + d1 * d1 + d2 * d2);
    float per = rot_err + trans_err;           // ROT_W = TRANS_W = 1
    float contrib = (cnt[bs] >= 6) ? 1.0f : 0.0f;

    float val = per * contrib, csum = contrib;
#pragma unroll
    for (int off = 16; off; off >>= 1) {
        val  += __shfl_down(val, off, 32);
        csum += __shfl_down(csum, off, 32);
    }
    if (bs == 0) out[0] = val / fmaxf(csum, 1.0f);
}

// =====================================================================
extern "C" void kernel_launch(void* const* d_in, const int* in_sizes, int n_in,
                              void* d_out, int out_size, void* d_ws, size_t ws_size,
                              hipStream_t stream)
{
    (void)in_sizes; (void)n_in; (void)out_size; (void)ws_size;
    // inputs: 0 dense_flow (unused), 1 dense_b, 2 dense_weight, 3 patch_cls,
    //         4 K, 5 poses, 6 gt_pose, 7 template_depth
    const float* db    = (const float*)d_in[1];
    const float* dw    = (const float*)d_in[2];
    const int*   pc    = (const int*)  d_in[3];
    const float* Kmat  = (const float*)d_in[4];
    const float* poses = (const float*)d_in[5];
    const float* gt    = (const float*)d_in[6];
    const float* depth = (const float*)d_in[7];
    float* out = (float*)d_out;

    char* ws = (char*)d_ws;
    float*  w_eff = (float*) (ws);                       // 32768 f  (128 KB)
    float4* XW    = (float4*)(ws + 131072);              // 32768 f4 (512 KB)
    int*    cnt   = (int*)   (ws + 655360);              // 32 i
    float*  Gw    = (float*) (ws + 655488);              // 32*256 f (32 KB)

    hipMemsetAsync(cnt, 0, 32 * sizeof(int), stream);

    const int NPATCH = Bn * Sn * Nq;                     // 32768
    k_wpatch<<<NPATCH / 8, 256, 0, stream>>>(db, dw, pc, w_eff, cnt);
    k_xobj  <<<NPATCH / 256, 256, 0, stream>>>(pc, Kmat, poses, depth, w_eff, XW);
    k_gram  <<<Bn * Sn, 256, 0, stream>>>(XW, Gw);
    k_solve <<<1, 32, 0, stream>>>(Gw, cnt, Kmat, gt, out);
}